// BoxEScorer_566935683334
// MI455X (gfx1250) — compile-verified
//
#include <hip/hip_runtime.h>
#include <hip/hip_bf16.h>
#include <math.h>

// Problem dims (fixed by the reference)
#define MM 1024   // x rows
#define NN 2048   // y rows
#define KK 128    // reduced dim
#define TJ 256    // threads per block = j-tile width
#define ITILE 16  // i rows per block
#define KC 4      // k's per staged chunk
#define NC (KK / KC)

// ---- CDNA5 async global->LDS path (guarded; sync fallback otherwise) ----
#if defined(__HIP_DEVICE_COMPILE__) && __has_builtin(__builtin_amdgcn_global_load_async_to_lds_b128)
#define HAS_ASYNC 1
#else
#define HAS_ASYNC 0
#endif

// The builtin's params are pointers to a 16B int vector (per hipcc diagnostic).
typedef int v4i __attribute__((vector_size(16)));
typedef __attribute__((address_space(1))) v4i g_v4i;   // global
typedef __attribute__((address_space(3))) v4i l_v4i;   // LDS

__device__ __forceinline__ void async_cp16(const void* g, const void* l) {
#if HAS_ASYNC
  __builtin_amdgcn_global_load_async_to_lds_b128(
      (g_v4i*)(unsigned long long)g,
      (l_v4i*)(unsigned)(unsigned long long)l,
      0, 0);
#else
  (void)g; (void)l;
#endif
}

__device__ __forceinline__ void wait_async_all() {
#if HAS_ASYNC
#if __has_builtin(__builtin_amdgcn_s_wait_asynccnt)
  __builtin_amdgcn_s_wait_asynccnt(0);
#else
  asm volatile("s_wait_asynccnt 0" ::: "memory");
#endif
#endif
}

// ---- per-element body: ~6 VALU ops ----
__device__ __forceinline__ void accum1(float xval, const float4& p, float hd, float& a) {
  float d  = xval - p.x;       // v_sub
  float l1 = fabsf(d);         // free source modifier on consumers
  bool  in = l1 <= hd;         // v_cmp -> vcc
  float m  = in ? p.z : p.y;   // v_cndmask (inv_dp1 : dp1)
  float b  = in ? 0.0f : p.w;  // v_cndmask (0 : c2)
  float t  = fmaf(l1, m, -b);  // v_fma
  a = fmaf(t, t, a);           // v_fma
}

__device__ __forceinline__ void issue_chunk(const float4* __restrict__ P,
                                            float4 (*buf)[TJ], int c, int jbase, int tid) {
#pragma unroll
  for (int r = 0; r < KC; ++r) {
    const float4* g = P + (size_t)(c * KC + r) * NN + jbase + tid;
#if HAS_ASYNC
    async_cp16((const void*)g, (const void*)&buf[r][tid]);
#else
    buf[r][tid] = *g;          // sync fallback: global load + ds_write
#endif
  }
}

// ---- Main: 16(i) x 256(j) output tile per block, double-buffered LDS ----
// Defined first so the disassembly snippet shows this kernel.
__global__ void __launch_bounds__(TJ)
boxe_main(const float4* __restrict__ P, const float* __restrict__ x,
          float* __restrict__ out) {
  __shared__ float4 Pbuf[2][KC][TJ];                 // 32 KB total LDS

  const int tid   = threadIdx.x;
  const int jbase = blockIdx.x * TJ;
  const int i0    = blockIdx.y * ITILE;
  const int j     = jbase + tid;

  // kick off chunk 0 immediately (async DMA overlaps accumulator setup)
  issue_chunk(P, Pbuf[0], 0, jbase, tid);

  float acc[ITILE];
#pragma unroll
  for (int ii = 0; ii < ITILE; ++ii) acc[ii] = 0.0f;

  for (int c = 0; c < NC; ++c) {
    // x addresses are block-uniform -> scalar (SMEM) loads, dual-issue with VALU
    float4 xc[ITILE];
#pragma unroll
    for (int ii = 0; ii < ITILE; ++ii)
      xc[ii] = *(const float4*)&x[(size_t)(i0 + ii) * KK + c * KC];

    wait_async_all();     // own async writes of chunk c landed in LDS
    __syncthreads();      // all waves landed chunk c; all done reading other buf
    if (c + 1 < NC) {
#if !HAS_ASYNC
      __builtin_prefetch(P + (size_t)(c + 1) * KC * NN + jbase + tid, 0, 1);
#endif
      issue_chunk(P, Pbuf[(c + 1) & 1], c + 1, jbase, tid);
    }

#pragma unroll
    for (int kk = 0; kk < KC; ++kk) {
      float4 p  = Pbuf[c & 1][kk][tid];     // ds_read_b128, conflict-free
      float  hd = fmaf(0.5f, p.y, -0.5f);   // half_delta = (dp1-1)/2
#pragma unroll
      for (int ii = 0; ii < ITILE; ++ii) {
        float xv = (kk == 0) ? xc[ii].x : (kk == 1) ? xc[ii].y
                 : (kk == 2) ? xc[ii].z : xc[ii].w;   // static select (kk unrolled)
        accum1(xv, p, hd, acc[ii]);
      }
    }
  }

#pragma unroll
  for (int ii = 0; ii < ITILE; ++ii)
    out[(size_t)(i0 + ii) * NN + j] = -sqrtf(acc[ii]);  // coalesced per i
}

// ---- Prep: per (j,k) params {center, dp1, 1/(dp1+eps), c2}, k-major ----
__global__ void boxe_prep(const float* __restrict__ y, float4* __restrict__ P) {
  int id = blockIdx.x * blockDim.x + threadIdx.x;   // 0 .. NN*KK-1
  int j = id & (NN - 1);
  int k = id >> 11;                                 // NN == 2048 == 2^11
  float mn  = y[j * (2 * KK) + k];
  float raw = y[j * (2 * KK) + KK + k];
  // stable softplus: max(r,0) + log1p(exp(-|r|))
  float sp = fmaxf(raw, 0.0f) + log1pf(expf(-fabsf(raw)));
  float center = fmaf(0.5f, sp, mn);
  float dp1 = sp + 1.0f;
  float inv = 1.0f / (dp1 + 1e-10f);
  float c2  = 0.5f * sp * (sp - 1.0f / (sp + 1e-10f));
  P[(size_t)k * NN + j] = make_float4(center, dp1, inv, c2);
}

// ---- Fallback if workspace too small: fully fused naive kernel ----
__global__ void boxe_naive(const float* __restrict__ y, const float* __restrict__ x,
                           float* __restrict__ out) {
  int id = blockIdx.x * blockDim.x + threadIdx.x;
  if (id >= MM * NN) return;
  int j = id % NN, i = id / NN;
  float acc = 0.0f;
  for (int k = 0; k < KK; ++k) {
    float mn  = y[j * (2 * KK) + k];
    float raw = y[j * (2 * KK) + KK + k];
    float sp  = fmaxf(raw, 0.0f) + log1pf(expf(-fabsf(raw)));
    float center = fmaf(0.5f, sp, mn);
    float dp1 = sp + 1.0f;
    float l1  = fabsf(x[i * KK + k] - center);
    float t   = (l1 <= 0.5f * sp)
                  ? l1 / (dp1 + 1e-10f)
                  : fmaf(l1, dp1, -0.5f * sp * (sp - 1.0f / (sp + 1e-10f)));
    acc = fmaf(t, t, acc);
  }
  out[id] = -sqrtf(acc);
}

extern "C" void kernel_launch(void* const* d_in, const int* in_sizes, int n_in,
                              void* d_out, int out_size, void* d_ws, size_t ws_size,
                              hipStream_t stream) {
  const float* y = (const float*)d_in[0];   // (2048, 256) f32
  const float* x = (const float*)d_in[1];   // (1024, 128) f32
  float* out = (float*)d_out;               // (1024, 2048) f32

  const size_t need = (size_t)KK * NN * sizeof(float4);   // 4 MB param table
  if (d_ws && ws_size >= need) {
    float4* P = (float4*)d_ws;
    boxe_prep<<<(NN * KK) / 256, 256, 0, stream>>>(y, P);
    dim3 grid(NN / TJ, MM / ITILE);                        // 8 x 64 = 512 blocks
    boxe_main<<<grid, TJ, 0, stream>>>(P, x, out);
  } else {
    boxe_naive<<<(MM * NN + 255) / 256, 256, 0, stream>>>(y, x, out);
  }
}